// Net_87668872446082
// MI455X (gfx1250) — compile-verified
//
#include <hip/hip_runtime.h>
#include <math.h>

typedef __attribute__((ext_vector_type(2))) float v2f;
typedef __attribute__((ext_vector_type(8))) float v8f;

#define WAVES 8

// ---------------------------------------------------------------------------
// Prologue: build M = U^T D U (16x16, symmetric) from thetas into d_ws.
// U = C23 R3 C23 C12 R2 C12 C01 R1 C01 R0 applied left-to-right from R0.
// 256 threads: thread t owns element (i,j) = (t>>4, t&15).
// ---------------------------------------------------------------------------
__global__ __launch_bounds__(256) void build_M_kernel(const float* __restrict__ thetas,
                                                      float* __restrict__ Mout) {
  __shared__ float U[256];
  __shared__ float th[4];
  const int t = threadIdx.x;
  const int i = t >> 4, j = t & 15;
  if (t < 4) th[t] = thetas[t];
  U[t] = (i == j) ? 1.0f : 0.0f;
  __syncthreads();

#define APPLY_RY(mask, theta) {                                        \
    float cc = cosf(0.5f * (theta));                                   \
    float ss = sinf(0.5f * (theta));                                   \
    int i0 = i & ~(mask); int i1 = i | (mask);                         \
    float v0 = U[i0 * 16 + j]; float v1 = U[i1 * 16 + j];              \
    float nv = (i & (mask)) ? fmaf(ss, v0, cc * v1)                    \
                            : fmaf(cc, v0, -ss * v1);                  \
    __syncthreads(); U[t] = nv; __syncthreads(); }

#define APPLY_CNOT(mc, mt) {                                           \
    int src = (i & (mc)) ? (i ^ (mt)) : i;                             \
    float nv = U[src * 16 + j];                                        \
    __syncthreads(); U[t] = nv; __syncthreads(); }

  // wire0 = bit3 (mask 8) ... wire3 = bit0 (mask 1)
  APPLY_RY(8, th[0]);
  APPLY_CNOT(8, 4);
  APPLY_RY(4, th[1]);
  APPLY_CNOT(8, 4);
  APPLY_CNOT(4, 2);
  APPLY_RY(2, th[2]);
  APPLY_CNOT(4, 2);
  APPLY_CNOT(2, 1);
  APPLY_RY(1, th[3]);
  APPLY_CNOT(2, 1);

  // M[i][j] = sum_k U[k][i] * Z[k] * U[k][j],  Z[k] = 4 - 2*popcount(k)
  float acc = 0.0f;
#pragma unroll
  for (int k = 0; k < 16; ++k) {
    float z = (float)(4 - 2 * __popc(k));
    acc = fmaf(U[k * 16 + i] * z, U[k * 16 + j], acc);
  }
  Mout[t] = acc;
}

// ---------------------------------------------------------------------------
// Main kernel: one wave32 per image. 49 patches -> 4 groups of 16 patches.
// Per group: lane L owns patch (L%16); Y = M (A-op) x Psi^T (B-op) via
// 4x V_WMMA_F32_16X16X4_F32. psi[4t+j] = w01[t]*w23[j], so WMMA operands and
// the quadratic-form factors are built from 8 scalars -- no arrays, no LDS.
// ---------------------------------------------------------------------------
__global__ __launch_bounds__(256) void qonv_kernel(const float* __restrict__ X,
                                                   const float* __restrict__ M,
                                                   const float* __restrict__ W,
                                                   const float* __restrict__ bias,
                                                   float* __restrict__ out,
                                                   int nImages) {
  __shared__ float cpix[WAVES][64];
  __shared__ float spix[WAVES][64];
  __shared__ float hbuf[WAVES][52];
  __shared__ float Wsh[500];  // [0..489] = W (10x49), [490..499] = bias

  const int tid  = threadIdx.x;
  const int wave = tid >> 5;
  const int lane = tid & 31;
  const int col  = tid & 15;        // lane % 16
  const int hi   = (tid >> 4) & 1;  // 1 for lanes 16..31

  for (int k = tid; k < 500; k += 256)
    Wsh[k] = (k < 490) ? W[k] : bias[k - 490];

  // Load M in A-operand layout (16x4 tiles): lane holds row m=col;
  // tile t: VGPR0 = M[m][4t + 2*hi], VGPR1 = M[m][4t + 2*hi + 1].
  v2f mA0, mA1, mA2, mA3;
  {
    const int base = col * 16 + 2 * hi;
    mA0.x = M[base +  0]; mA0.y = M[base +  1];
    mA1.x = M[base +  4]; mA1.y = M[base +  5];
    mA2.x = M[base +  8]; mA2.y = M[base +  9];
    mA3.x = M[base + 12]; mA3.y = M[base + 13];
  }

  const int img = blockIdx.x * WAVES + wave;
  const bool active = (img < nImages);

  // Stage per-pixel RY|0> components: cos = sqrt(1-x), sin = sqrt(x).
  if (active) {
    float x0 = X[img * 64 + lane];
    float x1 = X[img * 64 + lane + 32];
    cpix[wave][lane]      = sqrtf(fmaxf(1.0f - x0, 0.0f));
    spix[wave][lane]      = sqrtf(fmaxf(x0, 0.0f));
    cpix[wave][lane + 32] = sqrtf(fmaxf(1.0f - x1, 0.0f));
    spix[wave][lane + 32] = sqrtf(fmaxf(x1, 0.0f));
  }
  __syncthreads();

#pragma unroll
  for (int g = 0; g < 4; ++g) {
    const int p = g * 16 + col;            // patch index handled by this lane
    const bool valid = active && (p < 49);

    float c0 = 0, s0 = 0, c1 = 0, s1 = 0, c2 = 0, s2 = 0, c3 = 0, s3 = 0;
    if (valid) {
      int r = p / 7, cc = p % 7;
      int idx = r * 8 + cc;
      c0 = cpix[wave][idx];     s0 = spix[wave][idx];
      c1 = cpix[wave][idx + 1]; s1 = spix[wave][idx + 1];
      c2 = cpix[wave][idx + 8]; s2 = spix[wave][idx + 8];
      c3 = cpix[wave][idx + 9]; s3 = spix[wave][idx + 9];
    }
    // psi[4t+j] = w01[t] * w23[j]
    const float w01_0 = c0 * c1, w01_1 = c0 * s1, w01_2 = s0 * c1, w01_3 = s0 * s1;
    const float w23_0 = c2 * c3, w23_1 = c2 * s3, w23_2 = s2 * c3, w23_3 = s2 * s3;

    // B-operand tile t: lane = column (patch); rows K = 4t + {0,1} for
    // lanes 0-15, 4t + {2,3} for lanes 16-31  =>  w23 pair selected by hi.
    const float wx = hi ? w23_2 : w23_0;
    const float wy = hi ? w23_3 : w23_1;
    v2f b0, b1, b2, b3;
    b0.x = w01_0 * wx; b0.y = w01_0 * wy;
    b1.x = w01_1 * wx; b1.y = w01_1 * wy;
    b2.x = w01_2 * wx; b2.y = w01_2 * wy;
    b3.x = w01_3 * wx; b3.y = w01_3 * wy;

    v8f acc = {0, 0, 0, 0, 0, 0, 0, 0};
    acc = __builtin_amdgcn_wmma_f32_16x16x4_f32(false, mA0, false, b0, (short)0, acc, false, false);
    acc = __builtin_amdgcn_wmma_f32_16x16x4_f32(false, mA1, false, b1, (short)0, acc, false, false);
    acc = __builtin_amdgcn_wmma_f32_16x16x4_f32(false, mA2, false, b2, (short)0, acc, false, false);
    acc = __builtin_amdgcn_wmma_f32_16x16x4_f32(false, mA3, false, b3, (short)0, acc, false, false);

    // E = psi^T M psi : lane holds Y rows v + 8*hi of its own patch column.
    // psi[8*hi + v] = w01[2*hi + (v>=4)] * w23[v&3]
    const float uA = hi ? w01_2 : w01_0;
    const float uB = hi ? w01_3 : w01_1;
    float partial;
    partial = acc[0] * (uA * w23_0);
    partial = fmaf(acc[1], uA * w23_1, partial);
    partial = fmaf(acc[2], uA * w23_2, partial);
    partial = fmaf(acc[3], uA * w23_3, partial);
    partial = fmaf(acc[4], uB * w23_0, partial);
    partial = fmaf(acc[5], uB * w23_1, partial);
    partial = fmaf(acc[6], uB * w23_2, partial);
    partial = fmaf(acc[7], uB * w23_3, partial);

    float E = partial + __shfl_xor(partial, 16, 32);
    if (hi == 0 && p < 49) hbuf[wave][p] = fmaxf(E, 0.0f);
  }
  __syncthreads();

  // 49 -> 10 linear + log_softmax (lanes 0..9 own one logit each).
  const int o = (lane < 10) ? lane : 9;
  float logit = Wsh[490 + o];
#pragma unroll 7
  for (int p = 0; p < 49; ++p)
    logit = fmaf(hbuf[wave][p], Wsh[o * 49 + p], logit);

  float lv = (lane < 10) ? logit : -INFINITY;
  float mx = lv;
#pragma unroll
  for (int mask = 1; mask <= 8; mask <<= 1)
    mx = fmaxf(mx, __shfl_xor(mx, mask, 32));
  float e = (lane < 10) ? expf(logit - mx) : 0.0f;
  float se = e;
#pragma unroll
  for (int mask = 1; mask <= 8; mask <<= 1)
    se += __shfl_xor(se, mask, 32);
  float res = logit - mx - logf(se);

  if (active && lane < 10) out[img * 10 + lane] = res;
}

// ---------------------------------------------------------------------------
extern "C" void kernel_launch(void* const* d_in, const int* in_sizes, int n_in,
                              void* d_out, int out_size, void* d_ws, size_t ws_size,
                              hipStream_t stream) {
  const float* X      = (const float*)d_in[0];
  const float* thetas = (const float*)d_in[1];
  const float* W      = (const float*)d_in[2];
  const float* b      = (const float*)d_in[3];
  float* out = (float*)d_out;
  float* M   = (float*)d_ws;  // 256 floats = 1 KB

  const int nImages = in_sizes[0] / 64;

  build_M_kernel<<<1, 256, 0, stream>>>(thetas, M);

  const int blocks = (nImages + WAVES - 1) / WAVES;
  qonv_kernel<<<blocks, 256, 0, stream>>>(X, M, W, b, out, nImages);
}